// SymGatedGCNMambaOnlyModel_45844480918238
// MI455X (gfx1250) — compile-verified
//
#include <hip/hip_runtime.h>
#include <hip/hip_bf16.h>

// ---------------------------------------------------------------------------
// SymGatedGCN + Mamba forward for MI455X (gfx1250, wave32, WMMA).
// fp32 throughout (matches reference); dense GEMMs use V_WMMA_F32_16X16X4_F32.
// Workspace layout (floats), total ~56.7M floats (~227 MB):
//   h, x2pre, x2, ef, A1h..B2h, et(B3e), numf/denf/numb/denb, bn stats.
// ---------------------------------------------------------------------------

#define N_NODES 16384
#define N_EDGES 131072

typedef __attribute__((ext_vector_type(2))) float v2f;
typedef __attribute__((ext_vector_type(8))) float v8f;

__device__ __forceinline__ v8f wmma4(v2f a, v2f b, v8f c) {
  // D = A(16x4 f32) * B(4x16 f32) + C(16x16 f32)
  return __builtin_amdgcn_wmma_f32_16x16x4_f32(
      /*neg_a=*/false, a, /*neg_b=*/false, b,
      /*c_mod=*/(short)0, c, /*reuse_a=*/false, /*reuse_b=*/false);
}

__device__ __forceinline__ float sigmoidf_(float x) {
  return 1.f / (1.f + __expf(-x));
}

// --------------------------- small elementwise ------------------------------

__global__ void zero_kernel(float* __restrict__ p, size_t n) {
  for (size_t i = (size_t)blockIdx.x * blockDim.x + threadIdx.x; i < n;
       i += (size_t)gridDim.x * blockDim.x)
    p[i] = 0.f;
}

// h = l2n(relu(l1n(x)))  :  (n,2) -> (n,128)
__global__ void node_enc_kernel(const float* __restrict__ x,
                                const float* __restrict__ w1, const float* __restrict__ b1,
                                const float* __restrict__ w2, const float* __restrict__ b2,
                                float* __restrict__ h, int n) {
  size_t i = (size_t)blockIdx.x * blockDim.x + threadIdx.x;
  if (i >= (size_t)n * 128) return;
  int node = (int)(i >> 7), c = (int)(i & 127);
  float x0 = x[node * 2], x1 = x[node * 2 + 1];
  float acc = b2[c];
#pragma unroll
  for (int j = 0; j < 16; ++j) {
    float t = fmaxf(w1[j * 2] * x0 + w1[j * 2 + 1] * x1 + b1[j], 0.f);
    acc += w2[c * 16 + j] * t;
  }
  h[i] = acc;
}

// x2 = lbase(x2pre) : (n,4) -> (n,128)
__global__ void lbase_kernel(const float* __restrict__ xp,
                             const float* __restrict__ w, const float* __restrict__ b,
                             float* __restrict__ x2, int n) {
  size_t i = (size_t)blockIdx.x * blockDim.x + threadIdx.x;
  if (i >= (size_t)n * 128) return;
  int node = (int)(i >> 7), c = (int)(i & 127);
  float acc = b[c];
#pragma unroll
  for (int j = 0; j < 4; ++j) acc += w[c * 4 + j] * xp[node * 4 + j];
  x2[i] = acc;
}

// ------------------------------ Mamba scan ----------------------------------
// One wave32 per node.  lane = state index s (D_STATE=32); each lane holds
// h[d][s] for d=0..7 in registers.  Output taken at t = read_length-1 only.
__global__ void mamba_kernel(const float* __restrict__ reads, const int* __restrict__ rlen,
                             const float* __restrict__ inw,   // (16,4)
                             const float* __restrict__ cw,    // (8,4)
                             const float* __restrict__ cb,    // (8)
                             const float* __restrict__ xpw,   // (65,8)
                             const float* __restrict__ dtw,   // (8,1)
                             const float* __restrict__ dtb,   // (8)
                             const float* __restrict__ alog,  // (8,32)
                             const float* __restrict__ Dp,    // (8)
                             const float* __restrict__ opw,   // (4,8)
                             float* __restrict__ x2p, int n) {
  int wave = threadIdx.x >> 5, lane = threadIdx.x & 31;
  int node = blockIdx.x * 4 + wave;
  if (node >= n) return;
  int len = rlen[node];  // 1..64, wave-uniform

  float Aa[8];
#pragma unroll
  for (int d = 0; d < 8; ++d) Aa[d] = -__expf(alog[d * 32 + lane]);

  float hst[8];
  float w0_[8], w1_[8], w2_[8];  // conv window: xm[t-3], xm[t-2], xm[t-1]
#pragma unroll
  for (int d = 0; d < 8; ++d) { hst[d] = 0.f; w0_[d] = 0.f; w1_[d] = 0.f; w2_[d] = 0.f; }

  const float4* rp = (const float4*)(reads + (size_t)node * 256);
  const float* xrB = xpw + (1 + lane) * 8;
  const float* xrC = xpw + (33 + lane) * 8;

  for (int t = 0; t < len; ++t) {
    float4 r = rp[t];
    float xc[8];
#pragma unroll
    for (int d = 0; d < 8; ++d) {
      const float* iw = inw + d * 4;
      float xm = r.x * iw[0] + r.y * iw[1] + r.z * iw[2] + r.w * iw[3];
      float s = cw[d * 4 + 0] * w0_[d] + cw[d * 4 + 1] * w1_[d] +
                cw[d * 4 + 2] * w2_[d] + cw[d * 4 + 3] * xm + cb[d];
      xc[d] = s * sigmoidf_(s);  // silu
      w0_[d] = w1_[d]; w1_[d] = w2_[d]; w2_[d] = xm;
    }
    float dtr = 0.f, Bt = 0.f, Ct = 0.f;
#pragma unroll
    for (int j = 0; j < 8; ++j) {
      dtr += xc[j] * xpw[j];       // x_proj row 0 -> dt rank-1
      Bt  += xc[j] * xrB[j];       // B[s=lane]
      Ct  += xc[j] * xrC[j];       // C[s=lane]
    }
    bool last = (t == len - 1);
    float y8[8];
#pragma unroll
    for (int d = 0; d < 8; ++d) {
      float dpre = dtr * dtw[d] + dtb[d];
      float dt = (dpre > 20.f) ? dpre : log1pf(__expf(dpre));  // softplus
      float dA = __expf(dt * Aa[d]);
      hst[d] = hst[d] * dA + dt * Bt * xc[d];
      if (last) {
        float v = hst[d] * Ct;
#pragma unroll
        for (int off = 16; off > 0; off >>= 1) v += __shfl_xor(v, off, 32);
        y8[d] = v + xc[d] * Dp[d];
      }
    }
    if (last) {
#pragma unroll
      for (int d = 0; d < 8; ++d) {
        const float* iw = inw + (8 + d) * 4;
        float z = r.x * iw[0] + r.y * iw[1] + r.z * iw[2] + r.w * iw[3];
        y8[d] *= z * sigmoidf_(z);  // * silu(z)
      }
      if (lane < 4) {
        const float* orow = opw + lane * 8;
        float o = 0.f;
#pragma unroll
        for (int j = 0; j < 8; ++j) o += y8[j] * orow[j];
        x2p[node * 4 + lane] = o;
      }
    }
  }
}

// --------------------- fused edge encoder (WMMA, fp32) ----------------------
// 16 edges per block (128 threads = 4 waves).  Gather 16x512 into LDS, then
// layer1 (512->128, relu) and layer2 (128->128, relu) with WMMA f32 16x16x4.
__global__ void edge_mlp_kernel(const float* __restrict__ h, const float* __restrict__ x2,
                                const int* __restrict__ src, const int* __restrict__ dst,
                                const float* __restrict__ w1, const float* __restrict__ b1,
                                const float* __restrict__ w2, const float* __restrict__ b2,
                                float* __restrict__ ef, int E) {
  __shared__ float sdat[16 * 512];
  __shared__ float sint[16 * 128];
  int tid = threadIdx.x;
  int e0 = blockIdx.x * 16;

  for (int i = tid; i < 16 * 128; i += 128) {  // 128 float4 per edge row
    int e = i >> 7, c4 = i & 127;
    int seg = c4 >> 5, off = c4 & 31;
    int node = ((seg & 1) == 0) ? src[e0 + e] : dst[e0 + e];
    const float* bp = (seg < 2) ? (h + (size_t)node * 128) : (x2 + (size_t)node * 128);
    ((float4*)&sdat[e * 512 + seg * 128])[off] = ((const float4*)bp)[off];
  }
  __syncthreads();

  int wave = tid >> 5, lane = tid & 31;
  int m = lane & 15, kb = (lane >> 4) << 1;
  int nt0 = wave, nt1 = wave + 4;
  const float* wr0 = w1 + (size_t)(nt0 * 16 + m) * 512;
  const float* wr1 = w1 + (size_t)(nt1 * 16 + m) * 512;
  v8f acc0 = {}; v8f acc1 = {};
  for (int k = 0; k < 512; k += 4) {
    v2f a;  a[0] = sdat[m * 512 + k + kb];  a[1] = sdat[m * 512 + k + kb + 1];
    v2f bA; bA[0] = wr0[k + kb];            bA[1] = wr0[k + kb + 1];
    acc0 = wmma4(a, bA, acc0);
    v2f bB; bB[0] = wr1[k + kb];            bB[1] = wr1[k + kb + 1];
    acc1 = wmma4(a, bB, acc1);
  }
  {
    int nn = lane & 15, mrb = (lane >> 4) * 8;
#pragma unroll
    for (int v = 0; v < 8; ++v) {
      sint[(mrb + v) * 128 + nt0 * 16 + nn] = fmaxf(acc0[v] + b1[nt0 * 16 + nn], 0.f);
      sint[(mrb + v) * 128 + nt1 * 16 + nn] = fmaxf(acc1[v] + b1[nt1 * 16 + nn], 0.f);
    }
  }
  __syncthreads();

  const float* vr0 = w2 + (size_t)(nt0 * 16 + m) * 128;
  const float* vr1 = w2 + (size_t)(nt1 * 16 + m) * 128;
  v8f d0 = {}; v8f d1 = {};
  for (int k = 0; k < 128; k += 4) {
    v2f a;  a[0] = sint[m * 128 + k + kb];  a[1] = sint[m * 128 + k + kb + 1];
    v2f bA; bA[0] = vr0[k + kb];            bA[1] = vr0[k + kb + 1];
    d0 = wmma4(a, bA, d0);
    v2f bB; bB[0] = vr1[k + kb];            bB[1] = vr1[k + kb + 1];
    d1 = wmma4(a, bB, d1);
  }
  int nn = lane & 15, mrb = (lane >> 4) * 8;
#pragma unroll
  for (int v = 0; v < 8; ++v) {
    size_t row = (size_t)(e0 + mrb + v) * 128;
    ef[row + nt0 * 16 + nn] = fmaxf(d0[v] + b2[nt0 * 16 + nn], 0.f);
    ef[row + nt1 * 16 + nn] = fmaxf(d1[v] + b2[nt1 * 16 + nn], 0.f);
  }
}

// ------------------ generic WMMA GEMM: C = A @ W^T + bias -------------------
// grid (M/16, N/64), block 128 (4 waves, one 16x16 tile each).  K <= 512.
__global__ void gemm16_kernel(const float* __restrict__ A, const float* __restrict__ Wt,
                              const float* __restrict__ bias, float* __restrict__ C,
                              int K, int N, int do_relu) {
  __shared__ float sA[16 * 512];
  int tid = threadIdx.x;
  size_t row0 = (size_t)blockIdx.x * 16;
  int col0 = blockIdx.y * 64;
  int k4 = K >> 2;
  for (int i = tid; i < 16 * k4; i += 128) {
    int r = i / k4, cc = i % k4;
    ((float4*)&sA[r * K])[cc] = ((const float4*)(A + (row0 + r) * K))[cc];
  }
  __syncthreads();

  int wave = tid >> 5, lane = tid & 31;
  int m = lane & 15, kb = (lane >> 4) << 1;
  int ng0 = col0 + wave * 16;
  const float* wr = Wt + (size_t)(ng0 + m) * K;
  v8f acc = {};
  for (int k = 0; k < K; k += 4) {
    v2f a; a[0] = sA[m * K + k + kb]; a[1] = sA[m * K + k + kb + 1];
    v2f b; b[0] = wr[k + kb];         b[1] = wr[k + kb + 1];
    acc = wmma4(a, b, acc);
  }
  int nn = lane & 15, ng = ng0 + nn;
  float bs = bias[ng];
  int mrb = (lane >> 4) * 8;
#pragma unroll
  for (int v = 0; v < 8; ++v) {
    float val = acc[v] + bs;
    if (do_relu) val = fmaxf(val, 0.f);
    C[(row0 + mrb + v) * N + ng] = val;
  }
}

// ------------------------------ GCN fusions ---------------------------------

// et += B1h[src] + B2h[dst]; accumulate per-channel BN stats (LDS ds_add + atomics).
__global__ void edge_pre_bn_kernel(float* __restrict__ et,
                                   const float* __restrict__ B1h, const float* __restrict__ B2h,
                                   const int* __restrict__ src, const int* __restrict__ dst,
                                   float* __restrict__ esum, float* __restrict__ esq, int E) {
  __shared__ float ls[128], lq[128];
  int tid = threadIdx.x;
  if (tid < 128) { ls[tid] = 0.f; lq[tid] = 0.f; }
  __syncthreads();
  size_t base = (size_t)blockIdx.x * 8192;
  for (int j = tid; j < 8192; j += 256) {
    size_t i = base + j;
    int e = (int)(i >> 7), c = (int)(i & 127);
    float v = et[i] + B1h[(size_t)src[e] * 128 + c] + B2h[(size_t)dst[e] * 128 + c];
    et[i] = v;
    atomicAdd(&ls[c], v);
    atomicAdd(&lq[c], v * v);
  }
  __syncthreads();
  if (tid < 128) { atomicAdd(&esum[tid], ls[tid]); atomicAdd(&esq[tid], lq[tid]); }
}

// ef += relu(bn(et)); sigma = sigmoid(ef); scatter gated segment sums.
__global__ void edge_bn_gate_scatter_kernel(const float* __restrict__ et, float* __restrict__ ef,
                                            const float* __restrict__ A2h, const float* __restrict__ A3h,
                                            const int* __restrict__ src, const int* __restrict__ dst,
                                            const float* __restrict__ esum, const float* __restrict__ esq,
                                            const float* __restrict__ g, const float* __restrict__ bb,
                                            float* __restrict__ numf, float* __restrict__ denf,
                                            float* __restrict__ numb, float* __restrict__ denb, int E) {
  size_t i = (size_t)blockIdx.x * blockDim.x + threadIdx.x;
  if (i >= (size_t)E * 128) return;
  int e = (int)(i >> 7), c = (int)(i & 127);
  float inv = 1.f / (float)E;
  float mu = esum[c] * inv;
  float var = esq[c] * inv - mu * mu;
  float bn = (et[i] - mu) * rsqrtf(var + 1e-5f) * g[c] + bb[c];
  float en = ef[i] + fmaxf(bn, 0.f);
  ef[i] = en;
  float sg = sigmoidf_(en);
  int s = src[e], d = dst[e];
  atomicAdd(&numf[(size_t)d * 128 + c], A2h[(size_t)s * 128 + c] * sg);
  atomicAdd(&denf[(size_t)d * 128 + c], sg);
  atomicAdd(&numb[(size_t)s * 128 + c], A3h[(size_t)d * 128 + c] * sg);
  atomicAdd(&denb[(size_t)s * 128 + c], sg);
}

// hpre = A1h + numf/(denf+eps) + numb/(denb+eps); accumulate node BN stats.
__global__ void node_pre_bn_kernel(float* __restrict__ hpre,
                                   const float* __restrict__ numf, const float* __restrict__ denf,
                                   const float* __restrict__ numb, const float* __restrict__ denb,
                                   float* __restrict__ nsum, float* __restrict__ nsq, int n) {
  __shared__ float ls[128], lq[128];
  int tid = threadIdx.x;
  if (tid < 128) { ls[tid] = 0.f; lq[tid] = 0.f; }
  __syncthreads();
  size_t base = (size_t)blockIdx.x * 8192;
  for (int j = tid; j < 8192; j += 256) {
    size_t i = base + j;
    int c = (int)(i & 127);
    float v = hpre[i] + numf[i] / (denf[i] + 1e-6f) + numb[i] / (denb[i] + 1e-6f);
    hpre[i] = v;
    atomicAdd(&ls[c], v);
    atomicAdd(&lq[c], v * v);
  }
  __syncthreads();
  if (tid < 128) { atomicAdd(&nsum[tid], ls[tid]); atomicAdd(&nsq[tid], lq[tid]); }
}

// h += relu(bn(hpre))
__global__ void node_bn_res_kernel(float* __restrict__ h, const float* __restrict__ hpre,
                                   const float* __restrict__ nsum, const float* __restrict__ nsq,
                                   const float* __restrict__ g, const float* __restrict__ b, int n) {
  size_t i = (size_t)blockIdx.x * blockDim.x + threadIdx.x;
  if (i >= (size_t)n * 128) return;
  int c = (int)(i & 127);
  float inv = 1.f / (float)n;
  float mu = nsum[c] * inv;
  float var = nsq[c] * inv - mu * mu;
  float bn = (hpre[i] - mu) * rsqrtf(var + 1e-5f) * g[c] + b[c];
  h[i] += fmaxf(bn, 0.f);
}

// ------------------------- fused prediction head ----------------------------
// 16 edges/block: gather [h[src], h[dst], ef] (16x384) -> WMMA 384->64 relu ->
// 64->1 dot product.
__global__ void pred_kernel(const float* __restrict__ h, const float* __restrict__ ef,
                            const int* __restrict__ src, const int* __restrict__ dst,
                            const float* __restrict__ w1, const float* __restrict__ b1,
                            const float* __restrict__ w2, const float* __restrict__ b2,
                            float* __restrict__ out, int E) {
  __shared__ float sdat[16 * 384];
  __shared__ float sint[16 * 64];
  int tid = threadIdx.x;
  int e0 = blockIdx.x * 16;

  for (int i = tid; i < 16 * 96; i += 128) {  // 96 float4 per 384-float row
    int e = i / 96, c4 = i % 96;
    int seg = c4 >> 5, off = c4 & 31;
    const float* bp = (seg == 0) ? (h + (size_t)src[e0 + e] * 128)
                    : (seg == 1) ? (h + (size_t)dst[e0 + e] * 128)
                                 : (ef + (size_t)(e0 + e) * 128);
    ((float4*)&sdat[e * 384 + seg * 128])[off] = ((const float4*)bp)[off];
  }
  __syncthreads();

  int wave = tid >> 5, lane = tid & 31;
  int m = lane & 15, kb = (lane >> 4) << 1;
  const float* wr = w1 + (size_t)(wave * 16 + m) * 384;
  v8f acc = {};
  for (int k = 0; k < 384; k += 4) {
    v2f a; a[0] = sdat[m * 384 + k + kb]; a[1] = sdat[m * 384 + k + kb + 1];
    v2f b; b[0] = wr[k + kb];             b[1] = wr[k + kb + 1];
    acc = wmma4(a, b, acc);
  }
  int nn = lane & 15, ng = wave * 16 + nn, mrb = (lane >> 4) * 8;
#pragma unroll
  for (int v = 0; v < 8; ++v)
    sint[(mrb + v) * 64 + ng] = fmaxf(acc[v] + b1[ng], 0.f);
  __syncthreads();

  if (tid < 16) {
    float a = b2[0];
#pragma unroll
    for (int j = 0; j < 64; ++j) a += sint[tid * 64 + j] * w2[j];
    out[e0 + tid] = a;
  }
}

// ---------------------------------------------------------------------------

extern "C" void kernel_launch(void* const* d_in, const int* in_sizes, int n_in,
                              void* d_out, int out_size, void* d_ws, size_t ws_size,
                              hipStream_t stream) {
  (void)in_sizes; (void)n_in; (void)out_size; (void)ws_size;
  // setup_inputs() insertion order; params dict flattened in insertion order.
  const float* x          = (const float*)d_in[0];
  /* d_in[1] = e (EDGE_IN raw features) — unused by the reference forward */
  const float* reads      = (const float*)d_in[2];
  const int*   src        = (const int*)d_in[3];
  const int*   dst        = (const int*)d_in[4];
  const int*   rlen       = (const int*)d_in[5];
  const float* l1n_w      = (const float*)d_in[6];
  const float* l1n_b      = (const float*)d_in[7];
  const float* l2n_w      = (const float*)d_in[8];
  const float* l2n_b      = (const float*)d_in[9];
  const float* l1e_w      = (const float*)d_in[10];
  const float* l1e_b      = (const float*)d_in[11];
  const float* l2e_w      = (const float*)d_in[12];
  const float* l2e_b      = (const float*)d_in[13];
  const float* lbase_w    = (const float*)d_in[14];
  const float* lbase_b    = (const float*)d_in[15];
  const float* in_proj_w  = (const float*)d_in[16];
  const float* conv_w     = (const float*)d_in[17];
  const float* conv_b     = (const float*)d_in[18];
  const float* x_proj_w   = (const float*)d_in[19];
  const float* dt_proj_w  = (const float*)d_in[20];
  const float* dt_proj_b  = (const float*)d_in[21];
  const float* A_log      = (const float*)d_in[22];
  const float* Dp         = (const float*)d_in[23];
  const float* out_proj_w = (const float*)d_in[24];
  const float* gnn_w      = (const float*)d_in[25];
  const float* gnn_b      = (const float*)d_in[26];
  const float* bn_g       = (const float*)d_in[27];
  const float* bn_b       = (const float*)d_in[28];
  const float* pred_w1    = (const float*)d_in[29];
  const float* pred_b1    = (const float*)d_in[30];
  const float* pred_w2    = (const float*)d_in[31];
  const float* pred_b2    = (const float*)d_in[32];
  float* out = (float*)d_out;

  const size_t NN = N_NODES, NE = N_EDGES;
  float* W    = (float*)d_ws;
  float* h    = W;
  float* x2p  = h + NN * 128;
  float* x2   = x2p + NN * 4;
  float* ef   = x2 + NN * 128;
  float* A1h  = ef + NE * 128;
  float* A2h  = A1h + NN * 128;
  float* A3h  = A2h + NN * 128;
  float* B1h  = A3h + NN * 128;
  float* B2h  = B1h + NN * 128;
  float* et   = B2h + NN * 128;   // holds B3e then et
  float* numf = et + NE * 128;
  float* denf = numf + NN * 128;
  float* numb = denf + NN * 128;
  float* denb = numb + NN * 128;
  float* esum = denb + NN * 128;  // 128
  float* esq  = esum + 128;
  float* nsum = esq + 128;
  float* nsq  = nsum + 128;

  // ---- encoders ----
  node_enc_kernel<<<(NN * 128) / 256, 256, 0, stream>>>(x, l1n_w, l1n_b, l2n_w, l2n_b, h, NN);
  mamba_kernel<<<NN / 4, 128, 0, stream>>>(reads, rlen, in_proj_w, conv_w, conv_b, x_proj_w,
                                           dt_proj_w, dt_proj_b, A_log, Dp, out_proj_w, x2p, NN);
  lbase_kernel<<<(NN * 128) / 256, 256, 0, stream>>>(x2p, lbase_w, lbase_b, x2, NN);
  edge_mlp_kernel<<<NE / 16, 128, 0, stream>>>(h, x2, src, dst, l1e_w, l1e_b, l2e_w, l2e_b, ef, NE);

  // ---- 4 GCN layers ----
  for (int l = 0; l < 4; ++l) {
    zero_kernel<<<2048, 256, 0, stream>>>(numf, 4 * NN * 128 + 512);
    const float* gw = gnn_w + (size_t)l * 6 * 128 * 128;
    const float* gb = gnn_b + (size_t)l * 6 * 128;
    // node linears A1h,A2h,A3h,B1h,B2h
    gemm16_kernel<<<dim3(NN / 16, 2), 128, 0, stream>>>(h, gw + 0 * 16384, gb + 0 * 128, A1h, 128, 128, 0);
    gemm16_kernel<<<dim3(NN / 16, 2), 128, 0, stream>>>(h, gw + 1 * 16384, gb + 1 * 128, A2h, 128, 128, 0);
    gemm16_kernel<<<dim3(NN / 16, 2), 128, 0, stream>>>(h, gw + 2 * 16384, gb + 2 * 128, A3h, 128, 128, 0);
    gemm16_kernel<<<dim3(NN / 16, 2), 128, 0, stream>>>(h, gw + 3 * 16384, gb + 3 * 128, B1h, 128, 128, 0);
    gemm16_kernel<<<dim3(NN / 16, 2), 128, 0, stream>>>(h, gw + 4 * 16384, gb + 4 * 128, B2h, 128, 128, 0);
    // edge linear B3e -> et
    gemm16_kernel<<<dim3(NE / 16, 2), 128, 0, stream>>>(ef, gw + 5 * 16384, gb + 5 * 128, et, 128, 128, 0);

    edge_pre_bn_kernel<<<(NE * 128) / 8192, 256, 0, stream>>>(et, B1h, B2h, src, dst, esum, esq, NE);
    edge_bn_gate_scatter_kernel<<<(NE * 128) / 256, 256, 0, stream>>>(
        et, ef, A2h, A3h, src, dst, esum, esq,
        bn_g + (size_t)(l * 2 + 1) * 128, bn_b + (size_t)(l * 2 + 1) * 128,
        numf, denf, numb, denb, NE);
    node_pre_bn_kernel<<<(NN * 128) / 8192, 256, 0, stream>>>(A1h, numf, denf, numb, denb, nsum, nsq, NN);
    node_bn_res_kernel<<<(NN * 128) / 256, 256, 0, stream>>>(
        h, A1h, nsum, nsq, bn_g + (size_t)(l * 2 + 0) * 128, bn_b + (size_t)(l * 2 + 0) * 128, NN);
  }

  // ---- prediction head ----
  pred_kernel<<<NE / 16, 128, 0, stream>>>(h, ef, src, dst, pred_w1, pred_b1, pred_w2, pred_b2, out, NE);
}